// SpatialEmbLoss_55430847922732
// MI455X (gfx1250) — compile-verified
//
#include <hip/hip_runtime.h>

// ---------------- problem constants ----------------
#define HH 512
#define WW 1024
#define HWPIX (HH*WW)          // 524288 pixels per batch
#define NCH 31                 // 2 emb + 1 sigma + 8 seeds + 20 class
#define NB_C 384               // class-lovasz error histogram bins over [0,1]
#define NB_H 96                // hinge error histogram bins over [0,2]

// ---------------- workspace layout (4-byte word offsets) ----------------
#define OFF_CNT 0      // [2][64] float
#define OFF_SX  128
#define OFF_SY  256
#define OFF_SS  384
#define OFF_CX  512
#define OFF_CY  640
#define OFF_SE  768
#define OFF_SM  896
#define OFF_VAR 1024
#define OFF_FGT 1152
#define OFF_TOT 1280   // [2][8] float  (sum seeds^2 over all pixels)
#define OFF_FGS 1296   // [2][8] float  (sum seeds^2 over fg pixels)
#define OFF_FGC 1312   // [2][8] float  (fg pixel count)
#define OFF_CH  1328                    // class hist: 20*NB_C*2 uints
#define CH_WORDS (20*NB_C*2)
#define OFF_HHIST (OFF_CH + CH_WORDS)   // hinge hist: 2*64*NB_H*2 uints
#define HH_WORDS (2*64*NB_H*2)
#define ZERO_WORDS (OFF_HHIST + HH_WORDS)

typedef __attribute__((ext_vector_type(2)))  float    v2f;
typedef __attribute__((ext_vector_type(8)))  float    v8f;

__device__ __forceinline__ float sigmoidf_(float x){ return 1.0f/(1.0f+__expf(-x)); }

// ---------------- kernel 0: deterministic workspace clear ----------------
__global__ void k_zero(unsigned* ws, int n){
  int i = blockIdx.x*blockDim.x + threadIdx.x;
  if (i < n) ws[i] = 0u;
}

// ---------------- kernel 1: per-pixel stats + class lovasz histogram ----
__global__ __launch_bounds__(256) void k_pass1(const float* __restrict__ pred,
    const int* __restrict__ instp, const int* __restrict__ lblp,
    const float* __restrict__ xym, float* wsF, unsigned* wsU)
{
  const int b   = blockIdx.y;
  const int tid = threadIdx.x;
  __shared__ unsigned hC[20*NB_C*2];         // 60 KB
  __shared__ float cntL[64], sxL[64], syL[64], ssL[64];
  __shared__ float totL[8], fgsL[8], fgcL[8];
  for (int i=tid;i<20*NB_C*2;i+=256) hC[i]=0u;
  if (tid<64){ cntL[tid]=0.f; sxL[tid]=0.f; syL[tid]=0.f; ssL[tid]=0.f; }
  if (tid<8){ totL[tid]=0.f; fgsL[tid]=0.f; fgcL[tid]=0.f; }
  __syncthreads();

  const float* pb = pred + (size_t)b*NCH*HWPIX;
  const int*   lb = lblp + (size_t)b*HWPIX;
  const int*   ib = instp + (size_t)b*HWPIX;

  float aT[8], aF[8], aC[8];
  #pragma unroll
  for (int o=0;o<8;++o){ aT[o]=0.f; aF[o]=0.f; aC[o]=0.f; }

  const int TILES = HWPIX/256;   // 2048, exact
  for (int t = blockIdx.x; t < TILES; t += gridDim.x){
    int p  = t*256 + tid;
    int l  = lb[p], iv = ib[p];
    float xm = xym[p], ym = xym[HWPIX + p];
    float sg = pb[2*HWPIX + p];
    __builtin_prefetch(&pb[11*HWPIX + p]);   // class logits about to stream
    int m = (l>=12 && l<20 && iv>=1 && iv<=8) ? ((l-12)*8 + (iv-1)) : -1;
    if (m>=0){
      atomicAdd(&cntL[m],1.f); atomicAdd(&sxL[m],xm);
      atomicAdd(&syL[m],ym);   atomicAdd(&ssL[m],sg);
    }
    #pragma unroll
    for (int o=0;o<8;++o){
      float s = sigmoidf_(pb[(3+o)*HWPIX + p]);
      float q = s*s;
      aT[o] += q;
      if (l == 12+o){ aF[o]+=q; aC[o]+=1.f; }
    }
    // softmax over 20 class channels + error histogram
    float v[20]; float mx = -3.4e38f;
    #pragma unroll
    for (int c=0;c<20;++c){ v[c]=pb[(11+c)*HWPIX + p]; mx = fmaxf(mx, v[c]); }
    float ssum=0.f;
    #pragma unroll
    for (int c=0;c<20;++c){ v[c]=__expf(v[c]-mx); ssum+=v[c]; }
    float inv = 1.f/ssum;
    if (l != 0){
      #pragma unroll
      for (int c=0;c<20;++c){
        int fg = (l==c) ? 1 : 0;
        float e = fabsf((float)fg - v[c]*inv);
        int bin = (int)(e * NB_C); if (bin > NB_C-1) bin = NB_C-1;
        atomicAdd(&hC[(c*NB_C + bin)*2 + fg], 1u);
      }
    }
  }
  #pragma unroll
  for (int o=0;o<8;++o){
    atomicAdd(&totL[o], aT[o]); atomicAdd(&fgsL[o], aF[o]); atomicAdd(&fgcL[o], aC[o]);
  }
  __syncthreads();
  for (int i=tid;i<20*NB_C*2;i+=256){ unsigned u=hC[i]; if(u) atomicAdd(&wsU[OFF_CH+i],u); }
  if (tid<64){
    atomicAdd(&wsF[OFF_CNT + b*64 + tid], cntL[tid]);
    atomicAdd(&wsF[OFF_SX  + b*64 + tid], sxL[tid]);
    atomicAdd(&wsF[OFF_SY  + b*64 + tid], syL[tid]);
    atomicAdd(&wsF[OFF_SS  + b*64 + tid], ssL[tid]);
  }
  if (tid<8){
    atomicAdd(&wsF[OFF_TOT + b*8 + tid], totL[tid]);
    atomicAdd(&wsF[OFF_FGS + b*8 + tid], fgsL[tid]);
    atomicAdd(&wsF[OFF_FGC + b*8 + tid], fgcL[tid]);
  }
}

// ---------------- kernel 2: finalize centers / s_exp ----------------
__global__ void k_centers(float* wsF){
  int idx = threadIdx.x;   // 128 = 2 batches * 64 (o,k)
  if (idx < 128){
    float cc = fmaxf(wsF[OFF_CNT+idx], 1.f);
    float cx = wsF[OFF_SX+idx]/cc;
    float cy = wsF[OFF_SY+idx]/cc;
    float sm = wsF[OFF_SS+idx]/cc;
    wsF[OFF_CX+idx]=cx; wsF[OFF_CY+idx]=cy;
    wsF[OFF_SM+idx]=sm; wsF[OFF_SE+idx]=__expf(10.f*sm);
  }
}

// ---------------- kernel 3: WMMA dist matrix + hinge histograms ---------
// d2[p][n] = |e_p|^2 - 2 e_p . c_n + |c_n|^2 as a (16px x 4) x (4 x 16inst)
// f32 GEMM -> one V_WMMA_F32_16X16X4_F32 per 16x16 tile.
// Wave32 layout: A lanes 0-15 hold {K0,K1}={ex,ey} of row M=lane,
//                A lanes 16-31 hold {K2,K3}={|e|^2,1} of row M=lane-16.
//                B lanes 0-15 hold {K0,K1}={-2cx,-2cy} of col N=lane,
//                B lanes 16-31 hold {K2,K3}={1,|c|^2}  of col N=lane-16.
__global__ __launch_bounds__(256) void k_pass2(const float* __restrict__ pred,
    const int* __restrict__ instp, const int* __restrict__ lblp,
    const float* __restrict__ xym, float* wsF, unsigned* wsU)
{
  const int b    = blockIdx.y;
  const int tid  = threadIdx.x;
  const int lane = tid & 31;      // wave32
  const int wave = tid >> 5;
  __shared__ unsigned hH[64*NB_H*2];          // 48 KB hinge histogram
  __shared__ float cxL[64], cyL[64], seL[64], smL[64];
  __shared__ float varL[64], fgAL[64];
  __shared__ float exS[256], eyS[256];
  __shared__ int   mS[256];
  for (int i=tid;i<64*NB_H*2;i+=256) hH[i]=0u;
  if (tid<64){
    cxL[tid]=wsF[OFF_CX+b*64+tid];
    cyL[tid]=wsF[OFF_CY+b*64+tid];
    seL[tid]=wsF[OFF_SE+b*64+tid];
    smL[tid]=wsF[OFF_SM+b*64+tid];
    varL[tid]=0.f; fgAL[tid]=0.f;
  }
  __syncthreads();

  const float* pb = pred + (size_t)b*NCH*HWPIX;
  const int*   lb = lblp + (size_t)b*HWPIX;
  const int*   ib = instp + (size_t)b*HWPIX;

  const int TILES = HWPIX/256;
  for (int t = blockIdx.x; t < TILES; t += gridDim.x){
    int p  = t*256 + tid;
    int l  = lb[p], iv = ib[p];
    int m  = (l>=12 && l<20 && iv>=1 && iv<=8) ? ((l-12)*8 + (iv-1)) : -1;
    float ex = tanhf(pb[p])        + xym[p];
    float ey = tanhf(pb[HWPIX+p])  + xym[HWPIX+p];
    float sg = pb[2*HWPIX+p];
    exS[tid]=ex; eyS[tid]=ey; mS[tid]=m;
    if (m>=0){
      float dv = sg - smL[m];
      atomicAdd(&varL[m], dv*dv);                     // ds_add_f32
      float dx = ex - cxL[m], dy = ey - cyL[m];
      float dd = __expf(-seL[m]*(dx*dx+dy*dy));       // own-instance dist
      float sd = sigmoidf_(pb[(3 + (m>>3))*HWPIX + p]);
      float tt = sd - dd;
      atomicAdd(&fgAL[m], tt*tt);
    }
    __syncthreads();
    #pragma unroll
    for (int hpart=0; hpart<2; ++hpart){
      int base = wave*32 + hpart*16;   // 16-pixel tile staged in LDS
      int half = lane & 15;
      float aex = exS[base+half], aey = eyS[base+half];
      v2f A;
      if (lane < 16){ A[0]=aex; A[1]=aey; }
      else          { A[0]=aex*aex+aey*aey; A[1]=1.0f; }
      #pragma unroll
      for (int g=0; g<4; ++g){
        int instId = g*16 + half;
        float ccx = cxL[instId], ccy = cyL[instId];
        v2f Bm;
        if (lane < 16){ Bm[0]=-2.f*ccx; Bm[1]=-2.f*ccy; }
        else          { Bm[0]=1.0f;     Bm[1]=ccx*ccx+ccy*ccy; }
        v8f C = {};
        C = __builtin_amdgcn_wmma_f32_16x16x4_f32(
              false, A, false, Bm, (short)0, C, false, false);
        float sei = seL[instId];
        int rowOff = (lane < 16) ? 0 : 8;
        #pragma unroll
        for (int r=0;r<8;++r){
          float d2 = fmaxf(C[r], 0.f);
          float d  = __expf(-sei*d2);
          int pm   = mS[base + rowOff + r];
          int pos  = (pm == instId) ? 1 : 0;
          float e  = pos ? (2.f - 2.f*d) : (2.f*d);  // relu'd hinge errors
          int bin  = (int)(e * (NB_H*0.5f)); if (bin > NB_H-1) bin = NB_H-1;
          atomicAdd(&hH[(instId*NB_H + bin)*2 + pos], 1u);
        }
      }
    }
    __syncthreads();
  }
  if (tid<64){
    atomicAdd(&wsF[OFF_VAR + b*64 + tid], varL[tid]);
    atomicAdd(&wsF[OFF_FGT + b*64 + tid], fgAL[tid]);
  }
  for (int i=tid;i<64*NB_H*2;i+=256){
    unsigned u=hH[i]; if(u) atomicAdd(&wsU[OFF_HHIST + b*64*NB_H*2 + i], u);
  }
}

// ---------------- kernel 4: histogram-lovasz walks + final reduce -------
__global__ __launch_bounds__(256) void k_final(const float* wsF,
                                               const unsigned* wsU, float* out)
{
  __shared__ float hinge[128], pres[128];
  __shared__ float clsLoss[20], clsPres[20];
  int tid = threadIdx.x;
  if (tid < 128){           // one thread per (b,o,k) hinge row
    const unsigned* hb = wsU + OFF_HHIST + tid*NB_H*2;
    float P = 0.f;
    for (int bin=0;bin<NB_H;++bin) P += (float)hb[bin*2+1];
    float cp=0.f, cn=0.f, loss=0.f;
    float jprev = (P>0.f) ? 0.f : 1.f;           // J(0,0)
    for (int bin=NB_H-1; bin>=0; --bin){
      float pc=(float)hb[bin*2+1], nc=(float)hb[bin*2+0];
      if (pc+nc > 0.f){
        cp+=pc; cn+=nc;
        float jn = 1.f - (P-cp)/fmaxf(P+cn,1.f);
        loss += (bin+0.5f)*(2.0f/NB_H) * (jn - jprev);
        jprev = jn;
      }
    }
    hinge[tid]=loss; pres[tid]=(P>0.f)?1.f:0.f;
  }
  if (tid < 20){            // one thread per class (batches combined)
    const unsigned* hb = wsU + OFF_CH + tid*NB_C*2;
    float P=0.f;
    for (int bin=0;bin<NB_C;++bin) P += (float)hb[bin*2+1];
    float cp=0.f, cn=0.f, loss=0.f;
    float jprev=(P>0.f)?0.f:1.f;
    for (int bin=NB_C-1;bin>=0;--bin){
      float pc=(float)hb[bin*2+1], nc=(float)hb[bin*2+0];
      if (pc+nc>0.f){
        cp+=pc; cn+=nc;
        float jn = 1.f-(P-cp)/fmaxf(P+cn,1.f);
        loss += (bin+0.5f)*(1.0f/NB_C) * (jn-jprev);
        jprev=jn;
      }
    }
    clsLoss[tid]=loss; clsPres[tid]=(P>0.f)?1.f:0.f;
  }
  __syncthreads();
  if (tid==0){
    float np=0.f, cl=0.f;
    for (int c=0;c<20;++c) if (clsPres[c]>0.f){ cl+=clsLoss[c]; np+=1.f; }
    float class_loss = cl / fmaxf(np,1.f);
    float total = 0.f;
    for (int b=0;b<2;++b){
      float oc=0.f;
      for (int i=0;i<64;++i) oc += pres[b*64+i];
      float ocf = fmaxf(oc,1.f);
      float il=0.f, vl=0.f;
      for (int i=0;i<64;++i){
        int idx=b*64+i;
        if (pres[idx]>0.f){
          il += hinge[idx];
          vl += wsF[OFF_VAR+idx] / fmaxf(wsF[OFF_CNT+idx],1.f);
        }
      }
      il/=ocf; vl/=ocf;
      float sl=0.f;
      for (int o=0;o<8;++o)
        if (wsF[OFF_FGC + b*8 + o] > 0.f)
          sl += wsF[OFF_TOT+b*8+o] - wsF[OFF_FGS+b*8+o];
      for (int i=0;i<64;++i){ int idx=b*64+i; if (pres[idx]>0.f) sl += wsF[OFF_FGT+idx]; }
      sl /= (float)(HH*WW);
      total += 1.0f*il + 10.0f*vl + 1.0f*sl + class_loss;
    }
    out[0] = total*0.5f + 3.0f*class_loss;
  }
}

// ---------------- launcher ----------------
extern "C" void kernel_launch(void* const* d_in, const int* in_sizes, int n_in,
                              void* d_out, int out_size, void* d_ws, size_t ws_size,
                              hipStream_t stream) {
  (void)in_sizes; (void)n_in; (void)out_size; (void)ws_size;
  const float* pred  = (const float*)d_in[0];
  const int*   instp = (const int*)  d_in[1];
  const int*   lblp  = (const int*)  d_in[2];
  const float* xym   = (const float*)d_in[3];
  float*    wsF = (float*)d_ws;
  unsigned* wsU = (unsigned*)d_ws;
  float*    out = (float*)d_out;

  k_zero   <<<(ZERO_WORDS+255)/256, 256, 0, stream>>>(wsU, ZERO_WORDS);
  k_pass1  <<<dim3(256,2,1), 256, 0, stream>>>(pred, instp, lblp, xym, wsF, wsU);
  k_centers<<<1, 128, 0, stream>>>(wsF);
  k_pass2  <<<dim3(256,2,1), 256, 0, stream>>>(pred, instp, lblp, xym, wsF, wsU);
  k_final  <<<1, 256, 0, stream>>>(wsF, wsU, out);
}